// TransducerLoss_51994874085539
// MI455X (gfx1250) — compile-verified
//
#include <hip/hip_runtime.h>
#include <stdint.h>

// Problem constants (from the reference): B=16, MAXT=200, MAXU=101, A=512.
#define MAXT 200
#define MAXU 101
#define AV   512
#define CELLS (MAXT * MAXU)            // 20200 lattice cells per batch element
#define NDIAG ((MAXT - 1) + (MAXU - 1)) // 299 anti-diagonals after d=0

typedef __attribute__((ext_vector_type(4))) uint32_t uint32x4;
typedef __attribute__((ext_vector_type(8))) uint32_t uint32x8;

__device__ __forceinline__ float logaddexpf_(float a, float b) {
  float m = fmaxf(a, b);
  if (m == -__builtin_inff()) return m;      // both -inf
  return m + log1pf(expf(fminf(a, b) - m));  // matches jnp.logaddexp
}

// One workgroup per batch element. 128 threads (4 wave32) for the gather phase;
// wave 0 alone runs the register-resident anti-diagonal wavefront (no barriers).
// LDS layout (dynamic): blk[20200] | emit[20200] | lab[104]
extern "C" __global__ void __launch_bounds__(128)
rnnt_alpha_kernel(const float* __restrict__ lp, const int* __restrict__ labels,
                  const int* __restrict__ Tarr, const int* __restrict__ Uarr,
                  float* __restrict__ loss_ws) {
  extern __shared__ unsigned char smem[];
  float* blk_s  = (float*)smem;                 // CELLS floats
  float* emit_s = blk_s + CELLS;                // CELLS floats
  int*   lab_s  = (int*)(emit_s + CELLS);       // 104 ints (padded)

  const int b   = blockIdx.x;
  const int tid = threadIdx.x;
  const float* lp_b = lp + (size_t)b * (size_t)CELLS * (size_t)AV;

  // Labels (pad column MAXU-1 with blank=0, matching the reference's pad).
  if (tid < MAXU) lab_s[tid] = (tid < MAXU - 1) ? labels[b * (MAXU - 1) + tid] : 0;

  // ---- TDM: DMA the blank plane (log_probs[b,:,:,0], stride 512 elems) into LDS.
  // D# packed per CDNA5 ISA 8.3/8.4: 2D tile, 20200 rows x 1 elem, 4B elems.
  if (tid < 32) {  // wave 0 only (uniform per-wave branch); TDM ignores EXEC
    uint64_t ga  = (uint64_t)(uintptr_t)lp_b;           // global byte address of tile start
    uint32_t lds = (uint32_t)(uintptr_t)blk_s;          // low 32 bits = LDS byte offset
    uint32x4 g0 = {
      1u,                                               // count=1, is_restore=0, gather=0
      lds,                                              // lds_addr [63:32]
      (uint32_t)(ga & 0xFFFFFFFFu),                     // global_addr [95:64]
      (uint32_t)((ga >> 32) & 0x1FFFFFFu) | (2u << 30)  // global_addr[56:32] | type=2
    };
    uint32x8 g1 = {
      (2u << 16),                 // workgroup_mask=0, data_size=2 (4B), no flags
      (1u << 16),                 // atomic_barrier_addr=0 | tensor_dim0[15:0]=1
      ((uint32_t)CELLS) << 16,    // tensor_dim0[31:16]=0 | tensor_dim1[15:0]=20200
      (1u << 16),                 // tensor_dim1[31:16]=0 | tile_dim0=1
      (uint32_t)CELLS,            // tile_dim1=20200 | tile_dim2=0
      (uint32_t)AV,               // tensor_dim0_stride[31:0]=512 (elements)
      0u,                         // stride[47:32]=0 | tensor_dim1_stride[15:0]=0
      0u                          // tensor_dim1_stride[47:16]=0
    };
    asm volatile("tensor_load_to_lds %0, %1" :: "s"(g0), "s"(g1) : "memory");
    __builtin_amdgcn_s_wait_tensorcnt(0);   // TENSORcnt -> 0: LDS blank plane ready
  }
  __syncthreads();

  // ---- Gather emit plane: log_probs[b,t,u,lab[u]] -> LDS (all 4 waves).
  #pragma unroll 4
  for (int cell = tid; cell < CELLS; cell += 128) {
    int t = cell / MAXU;
    int u = cell - t * MAXU;
    emit_s[cell] = lp_b[(size_t)cell * AV + (size_t)lab_s[u]];
  }
  __syncthreads();

  if (tid >= 32) return;  // waves 1..3 retire; no further barriers

  // ---- Wave-0 register wavefront: lane L owns u in {4L..4L+3} (covers 0..127>=101).
  const float NEG  = -__builtin_inff();
  const int   lane = tid;
  const int   u0   = lane * 4;
  const int   Tb   = Tarr[b];
  const int   Ub   = Uarr[b];
  const int   dT   = (Tb - 1) + Ub;   // diagonal containing alpha[Tb-1, Ub]

  float prev[4], sv[4];
  #pragma unroll
  for (int j = 0; j < 4; ++j) { prev[j] = NEG; sv[j] = NEG; }
  if (lane == 0) prev[0] = 0.0f;      // alpha[0,0] = 0 (diag d=0)

  for (int d = 1; d <= NDIAG; ++d) {
    // prev[u0-1] lives in lane-1's prev[3]; lane 0's copy is unused (u=0 has no left).
    float pm1 = __shfl_up(prev[3], 1, 32);
    float cur[4];
    #pragma unroll
    for (int j = 0; j < 4; ++j) {
      const int  u  = u0 + j;
      const int  t  = d - u;
      const bool vA = (u <= MAXU - 1) &&
                      ((unsigned)(t - 1) <= (unsigned)(MAXT - 2)); // t in [1,199]
      const bool vB = (u >= 1) && (u <= MAXU - 1) &&
                      ((unsigned)t <= (unsigned)(MAXT - 1));       // t in [0,199]
      // Linear-in-d LDS indices: ia=(t-1)*101+u, ib=t*101+(u-1)=ia+100.
      const int ia = d * MAXU - 100 * u - MAXU;
      const int ib = ia + 100;
      float blkv = blk_s[ia];        // may be OOB -> value masked below (LDS never faults)
      float emv  = emit_s[ib];
      float pl   = (j == 0) ? pm1 : prev[j - 1];
      float at   = vA ? (prev[j] + blkv) : NEG;
      float bt   = vB ? (pl      + emv ) : NEG;
      cur[j] = logaddexpf_(at, bt);  // both NEG -> NEG: -inf propagates off-lattice
    }
    if (d == dT) {                   // uniform (Tb,Ub scalar): cheap scalar branch
      #pragma unroll
      for (int j = 0; j < 4; ++j) sv[j] = cur[j];
    }
    #pragma unroll
    for (int j = 0; j < 4; ++j) prev[j] = cur[j];
  }

  if (lane == (Ub >> 2)) {
    const int jj = Ub & 3;
    float av = (jj == 0) ? sv[0] : (jj == 1) ? sv[1] : (jj == 2) ? sv[2] : sv[3];
    // log_p = alpha[Tb-1, Ub] + blank at terminal cell; store -log_p
    loss_ws[b] = -(av + blk_s[(Tb - 1) * MAXU + Ub]);
  }
}

// Deterministic fixed-order mean (no atomics -> bit-identical across graph replays).
extern "C" __global__ void rnnt_reduce_kernel(const float* __restrict__ ws,
                                              float* __restrict__ out, int B) {
  if (threadIdx.x == 0 && blockIdx.x == 0) {
    float s = 0.0f;
    for (int i = 0; i < B; ++i) s += ws[i];
    out[0] = s / (float)B;
  }
}

extern "C" void kernel_launch(void* const* d_in, const int* in_sizes, int n_in,
                              void* d_out, int out_size, void* d_ws, size_t ws_size,
                              hipStream_t stream) {
  const float* lp     = (const float*)d_in[0];  // [B,200,101,512] f32
  const int*   labels = (const int*)  d_in[1];  // [B,100]
  const int*   Tarr   = (const int*)  d_in[2];  // [B]
  const int*   Uarr   = (const int*)  d_in[3];  // [B]
  const int B = in_sizes[2];

  float* ws = (float*)d_ws;  // B floats of per-utterance -log_p

  // Dynamic LDS: 2 planes + labels. (163 KB of the WGP's 320 KB; 16 WGPs active.)
  const size_t shmem = (size_t)(2 * CELLS) * sizeof(float)   // 161600
                     + 104 * sizeof(int)                     // labels (padded)
                     + 32;                                   // slack

  rnnt_alpha_kernel<<<B, 128, shmem, stream>>>(lp, labels, Tarr, Uarr, ws);
  rnnt_reduce_kernel<<<1, 32, 0, stream>>>(ws, (float*)d_out, B);
}